// DALoss_43679817400833
// MI455X (gfx1250) — compile-verified
//
#include <hip/hip_runtime.h>
#include <hip/hip_bf16.h>

// ---------------------------------------------------------------------------
// SSD DALoss for MI455X (gfx1250, wave32).
// Roofline: plabel = 362MB dominates; ~half skipped via domain_label => ~8us
// HBM floor at 23.3 TB/s. Focal softmax is staged per-tile into LDS with
// GLOBAL_LOAD_ASYNC_TO_LDS_B128 (ASYNCcnt), hard-negative mining is an exact
// per-sample radix-select on float bits in LDS (top-k SUM is tie-invariant),
// final mean over 128 samples uses V_WMMA_F32_16X16X4_F32.
// ---------------------------------------------------------------------------

#define N_   128
#define C_   81
#define A_   8732
#define TILE 192                 // anchors per focal tile (81*192*4B = 62KB LDS)
#define NT   46                  // ceil(A_/TILE)
#define NA   (N_ * A_)

#if __has_builtin(__builtin_amdgcn_global_load_async_to_lds_b128)
#define HAVE_ASYNC 1
#else
#define HAVE_ASYNC 0
#warning "no global_load_async_to_lds_b128 builtin on this pass - plain LDS staging"
#endif

#if __has_builtin(__builtin_amdgcn_wmma_f32_16x16x4_f32)
#define HAVE_WMMA 1
#else
#define HAVE_WMMA 0
#warning "no wmma_f32_16x16x4_f32 builtin on this pass - scalar reduce fallback"
#endif

#define AS1 __attribute__((address_space(1)))
#define AS3 __attribute__((address_space(3)))

typedef int   gv4i __attribute__((vector_size(16)));   // matches builtin param
typedef float v2f __attribute__((ext_vector_type(2)));
typedef float v8f __attribute__((ext_vector_type(8)));

__device__ __forceinline__ void wait_async0() {
#if __has_builtin(__builtin_amdgcn_s_wait_asynccnt)
  __builtin_amdgcn_s_wait_asynccnt(0);
#else
  asm volatile("s_wait_asynccnt 0" ::: "memory");
#endif
}

__device__ __forceinline__ float huber(float d) {
  float ad = fabsf(d);
  return ad < 1.0f ? 0.5f * d * d : ad - 0.5f;
}

// ---------------------------------------------------------------------------
// Kernel 1: per-sample smooth-L1 (positives, source domain) + positive count.
// ---------------------------------------------------------------------------
__global__ __launch_bounds__(256) void k_sl1(
    const float* __restrict__ ploc, const float* __restrict__ gloc,
    const int* __restrict__ glabel, const int* __restrict__ domain,
    const float* __restrict__ dboxes,
    float* __restrict__ sl1_sum, int* __restrict__ pos_cnt) {
  __shared__ float redf[256];
  __shared__ int   redi[256];
  const int n = blockIdx.x, t = threadIdx.x;
  const bool src = (domain[n] == 0);
  const float* pl = ploc + (size_t)n * 4 * A_;
  const float* gl = gloc + (size_t)n * 4 * A_;
  const int*   gb = glabel + (size_t)n * A_;
  float acc = 0.0f;
  int   cnt = 0;
  for (int a = t; a < A_; a += 256) {
    const int lab = gb[a];
    if (lab > 0) {
      ++cnt;
      if (src) {
        const float dbx = dboxes[a],        dby = dboxes[A_ + a];
        const float dbw = dboxes[2*A_ + a], dbh = dboxes[3*A_ + a];
        const float vx = 10.0f * (gl[a]        - dbx) / dbw;
        const float vy = 10.0f * (gl[A_ + a]   - dby) / dbh;
        const float vw = 5.0f * __logf(gl[2*A_ + a] / dbw);
        const float vh = 5.0f * __logf(gl[3*A_ + a] / dbh);
        acc += huber(pl[a] - vx) + huber(pl[A_ + a] - vy) +
               huber(pl[2*A_ + a] - vw) + huber(pl[3*A_ + a] - vh);
      }
    }
  }
  redf[t] = acc; redi[t] = cnt;
  __syncthreads();
  for (int s = 128; s > 0; s >>= 1) {
    if (t < s) { redf[t] += redf[t + s]; redi[t] += redi[t + s]; }
    __syncthreads();
  }
  if (t == 0) { sl1_sum[n] = redf[0]; pos_cnt[n] = redi[0]; }
}

// ---------------------------------------------------------------------------
// Kernel 2: focal loss per anchor. Tile of plabel async-staged into LDS,
// two-pass softmax (max, sum-exp). Writes con_neg (positives zeroed) and a
// per-tile partial sum of con over positives. Skips domain!=0 samples.
// ---------------------------------------------------------------------------
__global__ __launch_bounds__(256) void k_focal(
    const float* __restrict__ plabel, const int* __restrict__ glabel,
    const int* __restrict__ domain,
    float* __restrict__ con_neg, float* __restrict__ pos_con_part) {
  __shared__ float tile[C_ * TILE];   // 62,208 B
  __shared__ float red[256];
  const int n = blockIdx.y;
  if (domain[n] != 0) return;         // uniform early exit (no barriers yet)
  const int t  = threadIdx.x;
  const int a0 = blockIdx.x * TILE;
  const int rem = A_ - a0;
  const float* gbase = plabel + (size_t)n * C_ * A_ + a0;

  if (rem >= TILE) {
#if HAVE_ASYNC
    // 16B chunks: 81 rows * 48 chunks = 3888 async b128 loads per block.
    for (int i = t; i < C_ * (TILE / 4); i += 256) {
      const int c = i / (TILE / 4);
      const int w = (i - c * (TILE / 4)) * 4;
      const float* gp = gbase + (size_t)c * A_ + w;
      float* lp = &tile[c * TILE + w];
      __builtin_amdgcn_global_load_async_to_lds_b128(
          (AS1 gv4i*)gp, (AS3 gv4i*)lp, 0, 0);
    }
    wait_async0();
#else
    for (int i = t; i < C_ * TILE; i += 256) {
      const int c = i / TILE;
      tile[i] = gbase[(size_t)c * A_ + (i - c * TILE)];
    }
#endif
  } else {
    for (int i = t; i < C_ * TILE; i += 256) {
      const int c = i / TILE, x = i - c * TILE;
      tile[i] = (x < rem) ? gbase[(size_t)c * A_ + x] : 0.0f;
    }
  }
  __syncthreads();

  float pcon = 0.0f;
  const int a = a0 + t;
  if (t < TILE && a < A_) {
    const int lab = glabel[(size_t)n * A_ + a];
    float m = -3.402823466e+38f, xl = 0.0f;
    for (int c = 0; c < C_; ++c) {           // pass 1: max + gather label logit
      const float v = tile[c * TILE + t];
      if (c == lab) xl = v;
      m = fmaxf(m, v);
    }
    float s = 0.0f;
    for (int c = 0; c < C_; ++c)             // pass 2: sum exp
      s += __expf(tile[c * TILE + t] - m);
    const float logpt = xl - m - __logf(s);
    const float pt = __expf(logpt);
    const float om = 1.0f - pt;
    const float con = -0.25f * om * om * logpt;  // alpha=0.25, gamma=2
    if (lab > 0) {
      pcon = con;
      con_neg[(size_t)n * A_ + a] = 0.0f;
    } else {
      // clamp tiny negative rounding so float bits stay monotonic for select
      con_neg[(size_t)n * A_ + a] = fmaxf(con, 0.0f);
    }
  }
  red[t] = pcon;
  __syncthreads();
  for (int s = 128; s > 0; s >>= 1) {
    if (t < s) red[t] += red[t + s];
    __syncthreads();
  }
  if (t == 0) pos_con_part[(size_t)n * NT + blockIdx.x] = red[0];
}

// ---------------------------------------------------------------------------
// Kernel 3: per-sample exact radix-select of k = min(3*pos, A) largest
// con_neg values (sum only -- tie-order invariant), then per-sample loss.
// ---------------------------------------------------------------------------
__global__ __launch_bounds__(256) void k_select(
    const float* __restrict__ con_neg, const int* __restrict__ domain,
    const int* __restrict__ pos_cnt, const float* __restrict__ sl1_sum,
    const float* __restrict__ pos_con_part, float* __restrict__ sample_loss) {
  __shared__ unsigned int sv[A_];     // 34,928 B
  __shared__ int hist[256];
  __shared__ float redf[256];
  __shared__ unsigned int s_prefix;
  __shared__ int s_kr;
  const int n = blockIdx.x, t = threadIdx.x;
  if (domain[n] != 0 || pos_cnt[n] == 0) {
    if (t == 0) sample_loss[n] = 0.0f;
    return;
  }
  const int pos = pos_cnt[n];
  int k = 3 * pos; if (k > A_) k = A_;

  for (int a = t; a < A_; a += 256)
    sv[a] = __float_as_uint(con_neg[(size_t)n * A_ + a]);
  __syncthreads();

  unsigned int prefix = 0;
  int kr = k;
  for (int shift = 24; shift >= 0; shift -= 8) {
    hist[t] = 0;
    __syncthreads();
    const unsigned int hmask = (shift == 24) ? 0u : (0xFFFFFFFFu << (shift + 8));
    for (int a = t; a < A_; a += 256) {
      const unsigned int v = sv[a];
      if ((v & hmask) == prefix) atomicAdd(&hist[(v >> shift) & 0xFF], 1);
    }
    __syncthreads();
    if (t == 0) {
      int cum = 0, b = 255;
      for (; b > 0; --b) {
        const int h = hist[b];
        if (cum + h >= kr) break;
        cum += h;
      }
      s_prefix = prefix | ((unsigned int)b << shift);
      s_kr = kr - cum;
    }
    __syncthreads();
    prefix = s_prefix;
    kr = s_kr;
    __syncthreads();
  }
  // prefix == bits of the k-th largest value v*; kr = # of ties to include.
  const float vstar = __uint_as_float(prefix);
  float acc = 0.0f;
  for (int a = t; a < A_; a += 256) {
    const float v = __uint_as_float(sv[a]);
    if (v > vstar) acc += v;
  }
  redf[t] = acc;
  __syncthreads();
  for (int s = 128; s > 0; s >>= 1) {
    if (t < s) redf[t] += redf[t + s];
    __syncthreads();
  }
  if (t == 0) {
    float pcon = 0.0f;
    for (int i = 0; i < NT; ++i) pcon += pos_con_part[(size_t)n * NT + i];
    const float topk = redf[0] + (float)kr * vstar;
    sample_loss[n] = (sl1_sum[n] + pcon + topk) / (float)pos;
  }
}

// ---------------------------------------------------------------------------
// Kernel 4: mean over 128 per-sample losses via V_WMMA_F32_16X16X4_F32.
// Two chained WMMAs vs a ones B-matrix: sum(D) == 16 * sum(all 128 inputs).
// One wave, EXEC all ones.
// ---------------------------------------------------------------------------
__global__ __launch_bounds__(32) void k_final(
    const float* __restrict__ sample_loss, float* __restrict__ out) {
  const int lane = threadIdx.x;
  float s;
#if HAVE_WMMA
  v8f d = {};
  v2f bm; bm.x = 1.0f; bm.y = 1.0f;
  v2f am;
  am.x = sample_loss[lane * 2];
  am.y = sample_loss[lane * 2 + 1];
  d = __builtin_amdgcn_wmma_f32_16x16x4_f32(false, am, false, bm, (short)0, d,
                                            false, false);
  am.x = sample_loss[64 + lane * 2];
  am.y = sample_loss[64 + lane * 2 + 1];
  d = __builtin_amdgcn_wmma_f32_16x16x4_f32(false, am, false, bm, (short)0, d,
                                            false, false);
  s = (d[0] + d[1] + d[2] + d[3] + d[4] + d[5] + d[6] + d[7]) * (1.0f / 16.0f);
#else
  s = sample_loss[lane] + sample_loss[lane + 32] + sample_loss[lane + 64] +
      sample_loss[lane + 96];
#endif
  for (int off = 16; off > 0; off >>= 1) s += __shfl_xor(s, off, 32);
  if (lane == 0) out[0] = s * (1.0f / 128.0f);
}

// ---------------------------------------------------------------------------
extern "C" void kernel_launch(void* const* d_in, const int* in_sizes, int n_in,
                              void* d_out, int out_size, void* d_ws,
                              size_t ws_size, hipStream_t stream) {
  const float* ploc   = (const float*)d_in[0];
  const float* plabel = (const float*)d_in[1];
  const float* gloc   = (const float*)d_in[2];
  const int*   glabel = (const int*)d_in[3];
  const int*   domain = (const int*)d_in[4];
  const float* dboxes = (const float*)d_in[5];
  float* out = (float*)d_out;

  // Workspace layout (floats): con_neg[NA] | sl1[128] | sample[128] |
  // pos_con_part[128*NT] | pos_cnt[128] (as int). Total ~4.3 MB.
  float* ws           = (float*)d_ws;
  float* con_neg      = ws;
  float* sl1_sum      = ws + NA;
  float* sample_loss  = ws + NA + 128;
  float* pos_con_part = ws + NA + 256;
  int*   pos_cnt      = (int*)(ws + NA + 256 + N_ * NT);

  k_sl1<<<N_, 256, 0, stream>>>(ploc, gloc, glabel, domain, dboxes, sl1_sum,
                                pos_cnt);
  k_focal<<<dim3(NT, N_), 256, 0, stream>>>(plabel, glabel, domain, con_neg,
                                            pos_con_part);
  k_select<<<N_, 256, 0, stream>>>(con_neg, domain, pos_cnt, sl1_sum,
                                   pos_con_part, sample_loss);
  k_final<<<1, 32, 0, stream>>>(sample_loss, out);
}